// MultiHeadAttention_51677046505756
// MI455X (gfx1250) — compile-verified
//
#include <hip/hip_runtime.h>
#include <hip/hip_bf16.h>

#define B_    2
#define S_    2048
#define H_    16
#define DH_   64
#define DM_   1024
#define ROWS_ (B_ * S_)   // 4096

typedef __attribute__((ext_vector_type(2))) float v2f;
typedef __attribute__((ext_vector_type(8))) float v8f;

static __device__ __forceinline__ v2f ld2(const float* p) {
    return *reinterpret_cast<const v2f*>(p);
}

// D = A(16x4,f32) * B(4x16,f32) + C(16x16,f32) -> v_wmma_f32_16x16x4_f32
static __device__ __forceinline__ v8f wmma4(v2f a, v2f b, v8f c) {
    return __builtin_amdgcn_wmma_f32_16x16x4_f32(false, a, false, b, (short)0, c,
                                                 false, false);
}

// ---------------------------------------------------------------------------
// GEMM + bias: out = X[4096,1024] * W[1024,1024] + bias
// split_heads=1 -> out laid out [B,H,S,64]; else plain [4096,1024].
// 4 waves/block, each owns a 16x16 C tile.
// B chunk [64x16] double-buffered in LDS; A fragments batched (16 loads in
// flight) ahead of the WMMA chain so waits stagger instead of serializing.
// ---------------------------------------------------------------------------
__global__ __launch_bounds__(128)
void gemm_bias_kernel(const float* __restrict__ X, const float* __restrict__ W,
                      const float* __restrict__ bias, float* __restrict__ out,
                      int split_heads) {
    __shared__ float Bs[2][64][16];
    const int t    = threadIdx.x;
    const int w    = t >> 5;
    const int lane = t & 31;
    const int mrow = lane & 15;   // M row (A) / N col (B, C)
    const int khv  = lane >> 4;   // K half selector
    const int mbase = blockIdx.x * 64 + w * 16;
    const int n0    = blockIdx.y * 16;

    const int sr = t >> 3;        // staging row (0..15 per i-step base)
    const int sc = (t & 7) * 2;   // staging col pair

    // stage chunk 0 into buffer 0
    #pragma unroll
    for (int i = 0; i < 4; ++i) {
        int r  = sr + i * 16;
        v2f wv = ld2(&W[(size_t)r * DM_ + n0 + sc]);
        Bs[0][r][sc]     = wv.x;
        Bs[0][r][sc + 1] = wv.y;
    }

    v8f c   = {};
    int buf = 0;
    #pragma unroll 1
    for (int k0 = 0; k0 < DM_; k0 += 64) {
        __syncthreads();          // staged chunk `buf` is ready
        if (k0 + 64 < DM_) {      // issue next chunk's staging (overlaps WMMAs)
            #pragma unroll
            for (int i = 0; i < 4; ++i) {
                int r  = sr + i * 16;
                v2f wv = ld2(&W[(size_t)(k0 + 64 + r) * DM_ + n0 + sc]);
                Bs[buf ^ 1][r][sc]     = wv.x;
                Bs[buf ^ 1][r][sc + 1] = wv.y;
            }
        }
        // batch all 16 A fragments (keeps 16 b64 loads in flight)
        v2f af[16];
        #pragma unroll
        for (int kk = 0; kk < 16; ++kk)
            af[kk] = ld2(&X[(size_t)(mbase + mrow) * DM_ + k0 + kk * 4 + 2 * khv]);
        #pragma unroll
        for (int kk = 0; kk < 16; ++kk) {
            v2f b;
            b.x = Bs[buf][kk * 4 + 2 * khv][mrow];
            b.y = Bs[buf][kk * 4 + 2 * khv + 1][mrow];
            c = wmma4(af[kk], b, c);
        }
        buf ^= 1;
    }

    const int   ncol = n0 + mrow;
    const float bv   = bias[ncol];
    #pragma unroll
    for (int v = 0; v < 8; ++v) {
        int   row = mbase + v + 8 * khv;
        float val = c[v] + bv;
        if (split_heads) {
            int b = row >> 11;         // / S_
            int s = row & (S_ - 1);
            int h = ncol >> 6;
            int d = ncol & 63;
            out[((size_t)(b * H_ + h) * S_ + s) * DH_ + d] = val;
        } else {
            out[(size_t)row * DM_ + ncol] = val;
        }
    }
}

// ---------------------------------------------------------------------------
// Fused attention: scores (QK^T/8, mask) -> softmax -> write attn_prob ->
// P@V partial context, with probabilities held in registers throughout.
// Block: 8 waves = 256 threads, one (b,h) + 16 q-rows per block.
// Wave w owns k-columns [256w, 256w+256): 16 score tiles of 16x16.
// ---------------------------------------------------------------------------
__global__ __launch_bounds__(256)
void attn_fused_kernel(const float* __restrict__ qh, const float* __restrict__ kh,
                       const float* __restrict__ vh,
                       const unsigned char* __restrict__ mask,
                       float* __restrict__ P, float* __restrict__ ctx) {
    __shared__ float red[16][128];      // row-reduction partials
    __shared__ float redB[16][16];      // second-stage reduction
    __shared__ float rowmax[16];
    __shared__ float rowinv[16];
    __shared__ float tsc[8][16][18];    // per-wave C->A transpose scratch (padded)
    __shared__ float ctxbuf[8][16][64]; // per-wave partial context

    const int t    = threadIdx.x;
    const int w    = t >> 5;
    const int lane = t & 31;
    const int mrow = lane & 15;
    const int khv  = lane >> 4;
    const int bh   = blockIdx.y;
    const int bb   = bh >> 4;
    const int hh   = bh & 15;
    const int q0   = blockIdx.x * 16;
    const int kw   = w * 256;

    const float* qb = qh + (size_t)bh * S_ * DH_;
    const float* kb = kh + (size_t)bh * S_ * DH_;
    const float* Vb = vh + (size_t)bh * S_ * DH_;

    // Load the 16x64 q tile's A-fragments once (reused for all 16 k-tiles).
    v2f aq[16];
    #pragma unroll
    for (int kd = 0; kd < 16; ++kd)
        aq[kd] = ld2(&qb[(size_t)(q0 + mrow) * DH_ + kd * 4 + 2 * khv]);

    // ---- scores: 16 tiles of 16x16; batch the 16 K-fragments per tile ----
    v8f acc[16];
    #pragma unroll
    for (int nt = 0; nt < 16; ++nt) {
        const int n0 = kw + nt * 16;
        v2f bf[16];
        #pragma unroll
        for (int kd = 0; kd < 16; ++kd)
            bf[kd] = ld2(&kb[(size_t)(n0 + mrow) * DH_ + kd * 4 + 2 * khv]);
        v8f c = {};
        #pragma unroll
        for (int kd = 0; kd < 16; ++kd) c = wmma4(aq[kd], bf[kd], c);
        acc[nt] = c;
    }

    // ---- scale + mask fill ----
    #pragma unroll
    for (int nt = 0; nt < 16; ++nt) {
        const int kcol = kw + nt * 16 + mrow;
        #pragma unroll
        for (int v = 0; v < 8; ++v) {
            const int     qrow = q0 + v + 8 * khv;
            unsigned char m    = mask[(size_t)(bb * S_ + qrow) * S_ + kcol];
            acc[nt][v] = m ? -1e9f : acc[nt][v] * 0.125f;
        }
    }

    // ---- row max (deterministic LDS tree) ----
    {
        float pmax[8];
        #pragma unroll
        for (int v = 0; v < 8; ++v) pmax[v] = acc[0][v];
        #pragma unroll
        for (int nt = 1; nt < 16; ++nt)
            #pragma unroll
            for (int v = 0; v < 8; ++v) pmax[v] = fmaxf(pmax[v], acc[nt][v]);
        #pragma unroll
        for (int v = 0; v < 8; ++v) red[v + 8 * khv][w * 16 + mrow] = pmax[v];
    }
    __syncthreads();
    {
        const int r = t >> 4, i = t & 15;
        float m = red[r][i];
        #pragma unroll
        for (int j = 1; j < 8; ++j) m = fmaxf(m, red[r][i + 16 * j]);
        redB[r][i] = m;
    }
    __syncthreads();
    if (t < 16) {
        float m = redB[t][0];
        #pragma unroll
        for (int i = 1; i < 16; ++i) m = fmaxf(m, redB[t][i]);
        rowmax[t] = m;
    }
    __syncthreads();

    // ---- exp + row sum ----
    {
        float ps[8];
        #pragma unroll
        for (int v = 0; v < 8; ++v) ps[v] = 0.f;
        #pragma unroll
        for (int nt = 0; nt < 16; ++nt)
            #pragma unroll
            for (int v = 0; v < 8; ++v) {
                float e = __expf(acc[nt][v] - rowmax[v + 8 * khv]);
                acc[nt][v] = e;
                ps[v] += e;
            }
        #pragma unroll
        for (int v = 0; v < 8; ++v) red[v + 8 * khv][w * 16 + mrow] = ps[v];
    }
    __syncthreads();
    {
        const int r = t >> 4, i = t & 15;
        float s = red[r][i];
        #pragma unroll
        for (int j = 1; j < 8; ++j) s += red[r][i + 16 * j];
        redB[r][i] = s;
    }
    __syncthreads();
    if (t < 16) {
        float s = redB[t][0];
        #pragma unroll
        for (int i = 1; i < 16; ++i) s += redB[t][i];
        rowinv[t] = 1.0f / s;
    }
    __syncthreads();

    // ---- normalize + write attn_prob (single HBM write of P) ----
    #pragma unroll
    for (int nt = 0; nt < 16; ++nt) {
        const int kcol = kw + nt * 16 + mrow;
        #pragma unroll
        for (int v = 0; v < 8; ++v) {
            const int qrow = q0 + v + 8 * khv;
            float     pr   = acc[nt][v] * rowinv[v + 8 * khv];
            acc[nt][v] = pr;
            P[(size_t)(bh * S_ + qrow) * S_ + kcol] = pr;
        }
    }

    // ---- P@V: probs (regs) -> LDS transpose -> A fragments -> WMMA ----
    v8f cacc[4] = {};
    #pragma unroll
    for (int nt = 0; nt < 16; ++nt) {
        __syncthreads();   // protect previous iteration's reads of tsc
        #pragma unroll
        for (int v = 0; v < 8; ++v)
            tsc[w][v + 8 * khv][mrow] = acc[nt][v];   // [qrow][kcol] in LDS
        __syncthreads();

        // batch the 16 V B-fragments for this k-tile
        v2f vf[16];
        #pragma unroll
        for (int kk = 0; kk < 4; ++kk) {
            const int krow = kw + nt * 16 + kk * 4 + 2 * khv;
            #pragma unroll
            for (int dt = 0; dt < 4; ++dt) {
                v2f b;
                b.x = Vb[(size_t)krow * DH_ + dt * 16 + mrow];
                b.y = Vb[(size_t)(krow + 1) * DH_ + dt * 16 + mrow];
                vf[kk * 4 + dt] = b;
            }
        }
        // batch the 4 A fragments from the transpose scratch
        v2f at[4];
        #pragma unroll
        for (int kk = 0; kk < 4; ++kk) {
            at[kk].x = tsc[w][mrow][kk * 4 + 2 * khv];
            at[kk].y = tsc[w][mrow][kk * 4 + 2 * khv + 1];
        }
        #pragma unroll
        for (int kk = 0; kk < 4; ++kk)
            #pragma unroll
            for (int dt = 0; dt < 4; ++dt)
                cacc[dt] = wmma4(at[kk], vf[kk * 4 + dt], cacc[dt]);
    }

    // ---- deterministic cross-wave reduction of partial context ----
    #pragma unroll
    for (int dt = 0; dt < 4; ++dt)
        #pragma unroll
        for (int v = 0; v < 8; ++v)
            ctxbuf[w][v + 8 * khv][dt * 16 + mrow] = cacc[dt][v];
    __syncthreads();
    #pragma unroll
    for (int ii = 0; ii < 4; ++ii) {
        const int i = t + ii * 256;          // 0..1023 over [16 rows][64 cols]
        const int r = i >> 6, cidx = i & 63;
        float s = 0.f;
        #pragma unroll
        for (int ww = 0; ww < 8; ++ww) s += ctxbuf[ww][r][cidx];
        ctx[(size_t)(bb * S_ + q0 + r) * DM_ + hh * DH_ + cidx] = s;
    }
}

extern "C" void kernel_launch(void* const* d_in, const int* in_sizes, int n_in,
                              void* d_out, int out_size, void* d_ws, size_t ws_size,
                              hipStream_t stream) {
    (void)in_sizes; (void)n_in; (void)out_size; (void)ws_size;
    const float*         Q    = (const float*)d_in[0];
    const float*         K    = (const float*)d_in[1];
    const float*         V    = (const float*)d_in[2];
    const unsigned char* mask = (const unsigned char*)d_in[3];
    const float*         Wq   = (const float*)d_in[4];
    const float*         bq   = (const float*)d_in[5];
    const float*         Wk   = (const float*)d_in[6];
    const float*         bk   = (const float*)d_in[7];
    const float*         Wv   = (const float*)d_in[8];
    const float*         bv   = (const float*)d_in[9];
    const float*         Wo   = (const float*)d_in[10];
    const float*         bo   = (const float*)d_in[11];

    float* out = (float*)d_out;                    // output [4096,1024]
    float* P   = out + (size_t)ROWS_ * DM_;        // attn_prob [B,H,S,S]

    float* ws  = (float*)d_ws;                     // needs 64 MB
    float* qh  = ws;
    float* kh  = qh + (size_t)ROWS_ * DM_;
    float* vh  = kh + (size_t)ROWS_ * DM_;
    float* ctx = vh + (size_t)ROWS_ * DM_;

    dim3 gProj(ROWS_ / 64, DM_ / 16);
    gemm_bias_kernel<<<gProj, 128, 0, stream>>>(Q, Wq, bq, qh, 1);
    gemm_bias_kernel<<<gProj, 128, 0, stream>>>(K, Wk, bk, kh, 1);
    gemm_bias_kernel<<<gProj, 128, 0, stream>>>(V, Wv, bv, vh, 1);

    dim3 gAttn(S_ / 16, B_ * H_);
    attn_fused_kernel<<<gAttn, 256, 0, stream>>>(qh, kh, vh, mask, P, ctx);

    gemm_bias_kernel<<<gProj, 128, 0, stream>>>(ctx, Wo, bo, out, 0);
}